// CrossCompressUnit_7164005450408
// MI455X (gfx1250) — compile-verified
//
#include <hip/hip_runtime.h>

// Native clang vector types (needed for WMMA builtins and nontemporal b128 IO)
typedef float v2f __attribute__((ext_vector_type(2)));
typedef float v4f __attribute__((ext_vector_type(4)));
typedef float v8f __attribute__((ext_vector_type(8)));

#define DIM 64

__global__ __launch_bounds__(256) void cross_compress_wmma_kernel(
    const float* __restrict__ v, const float* __restrict__ e,
    const float* __restrict__ w_vv, const float* __restrict__ w_ev,
    const float* __restrict__ w_ve, const float* __restrict__ w_ee,
    const float* __restrict__ bias_v, const float* __restrict__ bias_e,
    float* __restrict__ v_out, float* __restrict__ e_out, int batch)
{
    // LDS: w_vv | w_ev | w_ve | w_ee | bias_v | bias_e  (6 x 64 floats)
    __shared__ __align__(16) float lds[6 * DIM];
    for (int t = threadIdx.x; t < 6 * DIM; t += 256) {
        const int which = t >> 6;
        const int o     = t & 63;
        const float* s = (which == 0) ? w_vv
                       : (which == 1) ? w_ev
                       : (which == 2) ? w_ve
                       : (which == 3) ? w_ee
                       : (which == 4) ? bias_v : bias_e;
        lds[t] = s[o];
    }
    __syncthreads();

    const int lane    = threadIdx.x & 31;
    const int wave    = threadIdx.x >> 5;
    const int m       = lane & 15;      // row within 16-row tile
    const int half    = lane >> 4;      // 0: cols 0-31, 1: cols 32-63
    const int colBase = half * 32;

    const int tile0 = (blockIdx.x * 8 + wave) * 16;   // wave-uniform
    if (tile0 >= batch) return;                        // uniform exit: EXEC stays full

    const int row   = tile0 + m;
    const int rload = (row < batch) ? row : (batch - 1);

    const v4f* __restrict__ vrow = reinterpret_cast<const v4f*>(v + (((size_t)rload) << 6) + colBase);
    const v4f* __restrict__ erow = reinterpret_cast<const v4f*>(e + (((size_t)rload) << 6) + colBase);

    // Stream this lane's 128B of v and 128B of e (8x b128 each)
    v4f v4[8], e4[8];
#pragma unroll
    for (int j = 0; j < 8; ++j) { v4[j] = vrow[j]; e4[j] = erow[j]; }

    // Per-lane partial dot products over this lane's 32 columns
    const v4f* Wvv = reinterpret_cast<const v4f*>(&lds[0 * DIM + colBase]);
    const v4f* Wev = reinterpret_cast<const v4f*>(&lds[1 * DIM + colBase]);
    const v4f* Wve = reinterpret_cast<const v4f*>(&lds[2 * DIM + colBase]);
    const v4f* Wee = reinterpret_cast<const v4f*>(&lds[3 * DIM + colBase]);

    float p_ev = 0.f, p_ee = 0.f, p_vv = 0.f, p_ve = 0.f;
#pragma unroll
    for (int j = 0; j < 8; ++j) {
        const v4f a = v4[j], b = e4[j];
        const v4f wev = Wev[j], wee = Wee[j], wvv = Wvv[j], wve = Wve[j];
        p_ev = fmaf(a.x, wev.x, fmaf(a.y, wev.y, fmaf(a.z, wev.z, fmaf(a.w, wev.w, p_ev))));
        p_ee = fmaf(a.x, wee.x, fmaf(a.y, wee.y, fmaf(a.z, wee.z, fmaf(a.w, wee.w, p_ee))));
        p_vv = fmaf(b.x, wvv.x, fmaf(b.y, wvv.y, fmaf(b.z, wvv.z, fmaf(b.w, wvv.w, p_vv))));
        p_ve = fmaf(b.x, wve.x, fmaf(b.y, wve.y, fmaf(b.z, wve.z, fmaf(b.w, wve.w, p_ve))));
    }

    // WMMA reduce + broadcast.
    // B (4x16, K x N): lane L<16 supplies (K=0,K=1) for column N=L via V0,V1;
    //                  lane L+16 supplies (K=2,K=3) for column N=L.
    // A (16x4) selector: rows {0,8} = [1,0,1,0]  -> D rows 0/8 = sum of K0+K2 partials
    //                    rows {1,9} = [0,1,0,1]  -> D rows 1/9 = sum of K1+K3 partials
    // D layout: lane L<16 reads (M=0..7, N=L) in V0..7; lane L+16 reads (M=8..15, N=L).
    // => every lane gets its OWN row's full dot products in D[0], D[1]. No shuffles.
    const float ax = ((m & 7) == 0) ? 1.0f : 0.0f;   // A[m][0] = A[m][2]
    const float ay = ((m & 7) == 1) ? 1.0f : 0.0f;   // A[m][1] = A[m][3]
    v2f A  = { ax, ay };
    v2f B1 = { p_ev, p_ee };   // dots of v against w_ev, w_ee
    v2f B2 = { p_vv, p_ve };   // dots of e against w_vv, w_ve
    v8f Z  = { 0.f, 0.f, 0.f, 0.f, 0.f, 0.f, 0.f, 0.f };

    v8f D1 = __builtin_amdgcn_wmma_f32_16x16x4_f32(false, A, false, B1, (short)0, Z, false, false);
    v8f D2 = __builtin_amdgcn_wmma_f32_16x16x4_f32(false, A, false, B2, (short)0, Z, false, false);

    const float d_ev = D1[0];   // v . w_ev   (for this lane's row)
    const float d_ee = D1[1];   // v . w_ee
    const float d_vv = D2[0];   // e . w_vv
    const float d_ve = D2[1];   // e . w_ve

    if (row < batch) {
        const v4f* BV = reinterpret_cast<const v4f*>(&lds[4 * DIM + colBase]);
        const v4f* BE = reinterpret_cast<const v4f*>(&lds[5 * DIM + colBase]);
        v4f* __restrict__ vo = reinterpret_cast<v4f*>(v_out + (((size_t)row) << 6) + colBase);
        v4f* __restrict__ eo = reinterpret_cast<v4f*>(e_out + (((size_t)row) << 6) + colBase);
#pragma unroll
        for (int j = 0; j < 8; ++j) {
            const v4f a = v4[j], b = e4[j], bv = BV[j], be = BE[j];
            v4f o1, o2;
            // v_out = v * (e.w_vv) + e * (v.w_ev) + bias_v
            o1.x = fmaf(a.x, d_vv, fmaf(b.x, d_ev, bv.x));
            o1.y = fmaf(a.y, d_vv, fmaf(b.y, d_ev, bv.y));
            o1.z = fmaf(a.z, d_vv, fmaf(b.z, d_ev, bv.z));
            o1.w = fmaf(a.w, d_vv, fmaf(b.w, d_ev, bv.w));
            // e_out = v * (e.w_ve) + e * (v.w_ee) + bias_e
            o2.x = fmaf(a.x, d_ve, fmaf(b.x, d_ee, be.x));
            o2.y = fmaf(a.y, d_ve, fmaf(b.y, d_ee, be.y));
            o2.z = fmaf(a.z, d_ve, fmaf(b.z, d_ee, be.z));
            o2.w = fmaf(a.w, d_ve, fmaf(b.w, d_ee, be.w));
            // Streaming outputs: non-temporal so they don't evict inputs from L2
            __builtin_nontemporal_store(o1, &vo[j]);
            __builtin_nontemporal_store(o2, &eo[j]);
        }
    }
}

extern "C" void kernel_launch(void* const* d_in, const int* in_sizes, int n_in,
                              void* d_out, int out_size, void* d_ws, size_t ws_size,
                              hipStream_t stream) {
    (void)n_in; (void)d_ws; (void)ws_size; (void)out_size;
    const float* v    = (const float*)d_in[0];
    const float* e    = (const float*)d_in[1];
    const float* w_vv = (const float*)d_in[2];
    const float* w_ev = (const float*)d_in[3];
    const float* w_ve = (const float*)d_in[4];
    const float* w_ee = (const float*)d_in[5];
    const float* b_v  = (const float*)d_in[6];
    const float* b_e  = (const float*)d_in[7];

    const int batch = in_sizes[0] / DIM;
    float* out   = (float*)d_out;
    float* v_out = out;
    float* e_out = out + (size_t)batch * DIM;

    const int rowsPerBlock = 128;  // 8 waves x 16 rows
    const int grid = (batch + rowsPerBlock - 1) / rowsPerBlock;
    cross_compress_wmma_kernel<<<grid, 256, 0, stream>>>(
        v, e, w_vv, w_ev, w_ve, w_ee, b_v, b_e, v_out, e_out, batch);
}